// NmpNet_19885698580924
// MI455X (gfx1250) — compile-verified
//
#include <hip/hip_runtime.h>

#define S_ 16
#define N_ 100
#define E_ (N_ * N_)
#define BPAD 40  // LDS row stride (dwords): (k+2)-row reads land 16 banks away -> no hi/lo conflict

typedef float v2f __attribute__((ext_vector_type(2)));
typedef float v8f __attribute__((ext_vector_type(8)));

__device__ __forceinline__ v8f wmma4(v2f a, v2f b, v8f c) {
    return __builtin_amdgcn_wmma_f32_16x16x4_f32(false, a, false, b, (short)0, c, false, false);
}

// ---------------- small helpers ----------------

__global__ void zero_stats_kernel(float* a, float* b) {
    int i = blockIdx.x * blockDim.x + threadIdx.x;
    if (i < S_ * 1024) { a[i] = 0.f; b[i] = 0.f; }
}

__global__ void finalize_stats_kernel(const float* __restrict__ sum, const float* __restrict__ sq,
                                      const float* __restrict__ g, const float* __restrict__ be,
                                      float* __restrict__ scale, float* __restrict__ shift,
                                      int Nout, float invR) {
    int i = blockIdx.x * blockDim.x + threadIdx.x;
    if (i >= S_ * Nout) return;
    int s = i / Nout, f = i - s * Nout;
    float m   = sum[s * 1024 + f] * invR;
    float var = sq[s * 1024 + f] * invR - m * m;
    float sc  = g[f] * rsqrtf(var + 1e-5f);
    scale[s * 1024 + f] = sc;
    shift[s * 1024 + f] = be[f] - m * sc;
}

// rel[e=(i,j)] = pos_j - pos_i, per scene
__global__ void rel_kernel(const float* __restrict__ pos, float* __restrict__ relB) {
    int idx = blockIdx.x * blockDim.x + threadIdx.x;  // S*E
    if (idx >= S_ * E_) return;
    int e = idx % E_, s = idx / E_;
    int i = e / N_, j = e - i * N_;
    relB[(size_t)idx * 2 + 0] = pos[(s * N_ + j) * 2 + 0] - pos[(s * N_ + i) * 2 + 0];
    relB[(size_t)idx * 2 + 1] = pos[(s * N_ + j) * 2 + 1] - pos[(s * N_ + i) * 2 + 1];
}

// fold spatial embedding through W1's last 64 rows: u,v,w0 s.t.
// rel_emb @ W1[128:192] = rx*u + ry*v + w0
__global__ void fold_spatial_kernel(const float* __restrict__ wsp, const float* __restrict__ bsp,
                                    const float* __restrict__ W1, float* __restrict__ uvw) {
    int col = blockIdx.x * blockDim.x + threadIdx.x;
    if (col >= 128) return;
    float u = 0.f, v = 0.f, w = 0.f;
    for (int c = 0; c < 64; ++c) {
        float wc = W1[(size_t)(128 + c) * 128 + col];
        u += wsp[c] * wc;
        v += wsp[64 + c] * wc;
        w += bsp[c] * wc;
    }
    uvw[col] = u; uvw[128 + col] = v; uvw[256 + col] = w;
}

// ---------------- fused GEMM + bias + per-scene stat accumulation ----------------
// MODE 0: A = [S][R][K], optional folded-BN affine (AFF)
// MODE 1: node2edge gather: row e=(i,j) -> [node_j | node_i] from [S][N][64], affine always on
// MODE 2: start gather: [x_j | x_i] from h_states, rank-1 rel term added in epilogue
// GUARD: true when R is not a multiple of 16 (partial last row tile needs masking)
// Block = 8 waves; each wave computes a 16x32 tile (2 accumulators, shared A fragment);
// B tile (K x 32) is staged in LDS once per block.
template <int MODE, bool AFF, bool GUARD>
__global__ void __launch_bounds__(256)
gemm_bn_kernel(const float* __restrict__ A,
               const float* __restrict__ scA, const float* __restrict__ shA,
               const float* __restrict__ W, const float* __restrict__ bias,
               const float* __restrict__ relB, const float* __restrict__ uvw,
               float* __restrict__ Z, float* __restrict__ sumB, float* __restrict__ sqB,
               int R, int K, int Nout) {
    __shared__ float Bs[128 * BPAD];
    const int tid = threadIdx.x;
    const int wave = tid >> 5, lane = tid & 31;
    const int lo = lane & 15, hi = lane >> 4;
    const int rowTiles = (R + 15) >> 4;
    const int rtBlocks = (rowTiles + 7) >> 3;
    const int s = blockIdx.x / rtBlocks;
    const int rtb = blockIdx.x - s * rtBlocks;
    const int rowBase = ((rtb << 3) + wave) << 4;
    const int colBase = blockIdx.y << 5;

    // stage B tile: W[k][colBase..colBase+31] -> Bs[k][c]
    for (int idx = tid; idx < K * 32; idx += 256) {
        int k = idx >> 5, c = idx & 31;
        Bs[k * BPAD + c] = W[(size_t)k * Nout + colBase + c];
    }
    __syncthreads();
    if (rowBase >= R) return;  // wave-uniform; no barriers after this

    const int row = rowBase + lo;
    const bool valid = GUARD ? (row < R) : true;
    const float msk = valid ? 1.f : 0.f;
    const int e = valid ? row : 0;

    v8f acc0 = {0.f, 0.f, 0.f, 0.f, 0.f, 0.f, 0.f, 0.f};
    v8f acc1 = {0.f, 0.f, 0.f, 0.f, 0.f, 0.f, 0.f, 0.f};
    const float* scRow = scA + s * 1024;
    const float* shRow = shA + s * 1024;

    if (MODE == 0) {
        const float* aRow = A + (size_t)(s * R + e) * K;
        for (int k0 = 0; k0 < K; k0 += 4) {
            const int kA = k0 + (hi << 1);  // lanes 0-15: K=k0,k0+1; 16-31: k0+2,k0+3
            v2f a = *(const v2f*)(aRow + kA);
            if (AFF) {
                v2f sc = *(const v2f*)(scRow + kA);
                v2f sh = *(const v2f*)(shRow + kA);
                a = a * sc + sh;
            }
            if (GUARD) a *= msk;
            v2f b0, b1;
            b0.x = Bs[kA * BPAD + lo];      b0.y = Bs[(kA + 1) * BPAD + lo];
            b1.x = Bs[kA * BPAD + 16 + lo]; b1.y = Bs[(kA + 1) * BPAD + 16 + lo];
            acc0 = wmma4(a, b0, acc0);
            acc1 = wmma4(a, b1, acc1);
        }
    } else {
        const int i = e / N_, j = e - i * N_;  // e = sender*N + receiver
        const float* p0 = A + ((size_t)s * N_ + j) * 64;  // k in [0,64)
        const float* p1 = A + ((size_t)s * N_ + i) * 64;  // k in [64,128)
#pragma unroll
        for (int half = 0; half < 2; ++half) {
            const float* p = half ? p1 : p0;
            for (int k0 = 0; k0 < 64; k0 += 4) {
                const int kA = k0 + (hi << 1);
                v2f a = *(const v2f*)(p + kA);
                if (MODE == 1) {  // affine over the 64 node features (same for both halves)
                    v2f sc = *(const v2f*)(scRow + kA);
                    v2f sh = *(const v2f*)(shRow + kA);
                    a = a * sc + sh;
                }
                if (GUARD) a *= msk;
                const int kB = (half << 6) + kA;
                v2f b0, b1;
                b0.x = Bs[kB * BPAD + lo];      b0.y = Bs[(kB + 1) * BPAD + lo];
                b1.x = Bs[kB * BPAD + 16 + lo]; b1.y = Bs[(kB + 1) * BPAD + 16 + lo];
                acc0 = wmma4(a, b0, acc0);
                acc1 = wmma4(a, b1, acc1);
            }
        }
    }

    // epilogue: bias (+rel rank-1 for MODE2), store z, accumulate per-scene stats
    const int col0 = colBase + lo, col1 = col0 + 16;
    float bc0 = bias[col0], bc1 = bias[col1];
    float u0 = 0.f, u1 = 0.f, vv0 = 0.f, vv1 = 0.f;
    if (MODE == 2) {
        u0 = uvw[col0];        u1 = uvw[col1];
        vv0 = uvw[128 + col0]; vv1 = uvw[128 + col1];
        bc0 += uvw[256 + col0];
        bc1 += uvw[256 + col1];
    }
    float ps0 = 0.f, pq0 = 0.f, ps1 = 0.f, pq1 = 0.f;
#pragma unroll
    for (int v = 0; v < 8; ++v) {
        int r = rowBase + v + (hi << 3);  // C layout: vgpr v -> M=v (lanes<16) / v+8
        if (!GUARD || r < R) {
            float z0 = acc0[v] + bc0;
            float z1 = acc1[v] + bc1;
            if (MODE == 2) {
                v2f rl = *(const v2f*)(relB + ((size_t)s * E_ + r) * 2);
                z0 += rl.x * u0 + rl.y * vv0;
                z1 += rl.x * u1 + rl.y * vv1;
            }
            size_t o = (size_t)(s * R + r) * Nout;
            Z[o + col0] = z0;
            Z[o + col1] = z1;
            ps0 += z0; pq0 += z0 * z0;
            ps1 += z1; pq1 += z1 * z1;
        }
    }
    atomicAdd(&sumB[s * 1024 + col0], ps0);
    atomicAdd(&sqB[s * 1024 + col0], pq0);
    atomicAdd(&sumB[s * 1024 + col1], ps1);
    atomicAdd(&sqB[s * 1024 + col1], pq1);
}

// edge2node with previous BN affine folded: nodeIn[s][j][c] = (sc*sum_i z + N*sh)/64
__global__ void edge2node_kernel(const float* __restrict__ zE, const float* __restrict__ sc,
                                 const float* __restrict__ sh, float* __restrict__ nodeIn) {
    int idx = blockIdx.x * blockDim.x + threadIdx.x;  // S*N*64
    if (idx >= S_ * N_ * 64) return;
    int c = idx & 63;
    int j = (idx >> 6) % N_;
    int s = idx / (N_ * 64);
    const float* base = zE + ((size_t)s * E_ + j) * 64 + c;
    float acc = 0.f;
    for (int i = 0; i < N_; ++i) acc += base[(size_t)i * N_ * 64];
    float scale = sc[s * 1024 + c], shift = sh[s * 1024 + c];
    nodeIn[idx] = (scale * acc + (float)N_ * shift) * (1.0f / 64.0f);
}

__global__ void apply_bn_relu_kernel(const float* __restrict__ z, const float* __restrict__ sc,
                                     const float* __restrict__ sh, float* __restrict__ out,
                                     int Nout, int total) {
    int idx = blockIdx.x * blockDim.x + threadIdx.x;
    if (idx >= total) return;
    int f = idx % Nout;
    int s = idx / (N_ * Nout);
    float v = z[idx] * sc[s * 1024 + f] + sh[s * 1024 + f];
    out[idx] = fmaxf(v, 0.f);
}

// ---------------- launcher ----------------

extern "C" void kernel_launch(void* const* d_in, const int* in_sizes, int n_in,
                              void* d_out, int out_size, void* d_ws, size_t ws_size,
                              hipStream_t stream) {
    auto in = [&](int i) { return (const float*)d_in[i]; };

    float* ws = (float*)d_ws;
    float* zE     = ws;                                 // [S][E][128]
    float* zE2    = zE + (size_t)S_ * E_ * 128;         // [S][E][64]
    float* nodeIn = zE2 + (size_t)S_ * E_ * 64;         // [S][N][64]
    float* nodeZ1 = nodeIn + S_ * N_ * 64;              // [S][N][128]
    float* nodeZ2 = nodeZ1 + S_ * N_ * 128;             // [S][N][64]
    float* nodeH1 = nodeZ2 + S_ * N_ * 64;              // [S][N][128]
    float* nodeZo = nodeH1 + S_ * N_ * 128;             // [S][N][1024]
    float* sumB = nodeZo + (size_t)S_ * N_ * 1024;      // [S][1024]
    float* sqB  = sumB + S_ * 1024;
    float* scA  = sqB + S_ * 1024;
    float* shA  = scA + S_ * 1024;
    float* scB  = shA + S_ * 1024;
    float* shB  = scB + S_ * 1024;
    float* relB = shB + S_ * 1024;                      // [S][E][2]
    float* uvw  = relB + (size_t)S_ * E_ * 2;           // [3][128]

    auto zero = [&]() {
        zero_stats_kernel<<<(S_ * 1024 + 255) / 256, 256, 0, stream>>>(sumB, sqB);
    };
    auto fin = [&](const float* g, const float* be, int Nout, float invR, float* sc, float* sh) {
        finalize_stats_kernel<<<(S_ * Nout + 255) / 256, 256, 0, stream>>>(
            sumB, sqB, g, be, sc, sh, Nout, invR);
    };

    const float* hst = in(0);
    const float* pos = in(1);
    const float* wsp = in(2);
    const float* bsp = in(3);
    dim3 blk(256);
    auto eGrid = [&](int Nout) { return dim3(S_ * ((625 + 7) / 8), Nout / 32); };
    auto nGrid = [&](int Nout) { return dim3(S_, Nout / 32); };

    // precompute rel buffer and folded spatial embedding
    rel_kernel<<<(S_ * E_ + 255) / 256, 256, 0, stream>>>(pos, relB);
    fold_spatial_kernel<<<1, 128, 0, stream>>>(wsp, bsp, in(4), uvw);

    // start MLP L1: [x_j|x_i](128)@W1[0:128] + rank-1 rel epilogue   (R=10000 -> no guard)
    zero();
    gemm_bn_kernel<2, false, false><<<eGrid(128), blk, 0, stream>>>(
        hst, scA, shA, in(4), in(5), relB, uvw, zE, sumB, sqB, E_, 128, 128);
    fin(in(6), in(7), 128, 1.f / E_, scA, shA);
    // start MLP L2: 128 -> 64, BN folded into A
    zero();
    gemm_bn_kernel<0, true, false><<<eGrid(64), blk, 0, stream>>>(
        zE, scA, shA, in(8), in(9), relB, uvw, zE2, sumB, sqB, E_, 128, 64);
    fin(in(10), in(11), 64, 1.f / E_, scB, shB);

    for (int t = 0; t < 3; ++t) {
        int b = 12 + 16 * t;
        edge2node_kernel<<<(S_ * N_ * 64 + 255) / 256, 256, 0, stream>>>(zE2, scB, shB, nodeIn);
        // node MLP L1: 64 -> 128   (R=100 -> guarded)
        zero();
        gemm_bn_kernel<0, false, true><<<nGrid(128), blk, 0, stream>>>(
            nodeIn, scA, shA, in(b + 0), in(b + 1), relB, uvw, nodeZ1, sumB, sqB, N_, 64, 128);
        fin(in(b + 2), in(b + 3), 128, 1.f / N_, scA, shA);
        // node MLP L2: 128 -> 64
        zero();
        gemm_bn_kernel<0, true, true><<<nGrid(64), blk, 0, stream>>>(
            nodeZ1, scA, shA, in(b + 4), in(b + 5), relB, uvw, nodeZ2, sumB, sqB, N_, 128, 64);
        fin(in(b + 6), in(b + 7), 64, 1.f / N_, scB, shB);
        // edge MLP L1: node2edge gather (128) -> 128
        zero();
        gemm_bn_kernel<1, true, false><<<eGrid(128), blk, 0, stream>>>(
            nodeZ2, scB, shB, in(b + 8), in(b + 9), relB, uvw, zE, sumB, sqB, E_, 128, 128);
        fin(in(b + 10), in(b + 11), 128, 1.f / E_, scA, shA);
        // edge MLP L2: 128 -> 64
        zero();
        gemm_bn_kernel<0, true, false><<<eGrid(64), blk, 0, stream>>>(
            zE, scA, shA, in(b + 12), in(b + 13), relB, uvw, zE2, sumB, sqB, E_, 128, 64);
        fin(in(b + 14), in(b + 15), 64, 1.f / E_, scB, shB);
    }

    // end MLP (ReLU after BN -> explicit apply)
    edge2node_kernel<<<(S_ * N_ * 64 + 255) / 256, 256, 0, stream>>>(zE2, scB, shB, nodeIn);
    zero();
    gemm_bn_kernel<0, false, true><<<nGrid(128), blk, 0, stream>>>(
        nodeIn, scA, shA, in(60), in(61), relB, uvw, nodeZ1, sumB, sqB, N_, 64, 128);
    fin(in(62), in(63), 128, 1.f / N_, scA, shA);
    apply_bn_relu_kernel<<<(S_ * N_ * 128 + 255) / 256, 256, 0, stream>>>(
        nodeZ1, scA, shA, nodeH1, 128, S_ * N_ * 128);
    zero();
    gemm_bn_kernel<0, false, true><<<nGrid(1024), blk, 0, stream>>>(
        nodeH1, scA, shA, in(64), in(65), relB, uvw, nodeZo, sumB, sqB, N_, 128, 1024);
    fin(in(66), in(67), 1024, 1.f / N_, scB, shB);
    apply_bn_relu_kernel<<<(S_ * N_ * 1024 + 255) / 256, 256, 0, stream>>>(
        nodeZo, scB, shB, (float*)d_out, 1024, S_ * N_ * 1024);
}